// BaseA3c_5617817223475
// MI455X (gfx1250) — compile-verified
//
#include <hip/hip_runtime.h>
#include <cstdint>
#include <cstddef>

#define GAMMA 0.99f
#define BETA  0.01f
#define T_LEN 128
#define A_DIM 8
#define ROWS  32            // batch rows per block
#define THREADS 256
#define STRIDE 129          // padded LDS row stride (floats) -> conflict-free column reads

typedef unsigned int u32x4 __attribute__((ext_vector_type(4)));
typedef int          i32x8 __attribute__((ext_vector_type(8)));
typedef int          i32x4 __attribute__((ext_vector_type(4)));
typedef float        v4f   __attribute__((ext_vector_type(4)));

__global__ __launch_bounds__(THREADS)
void a3c_scan_kernel(const float* __restrict__ values,
                     const float* __restrict__ last_value,
                     const float* __restrict__ rewards,
                     const float* __restrict__ log_probs,
                     const float* __restrict__ entropies,
                     const int*   __restrict__ terminal_mask,
                     float* __restrict__ out)
{
    __shared__ float smem[3 * ROWS * STRIDE + ROWS];
    float* v_s  = smem;                       // values tile  [ROWS][STRIDE]
    float* r_s  = smem + ROWS * STRIDE;       // rewards tile [ROWS][STRIDE]
    float* lp_s = smem + 2 * ROWS * STRIDE;   // sum_a log_probs [ROWS][STRIDE]
    float* es_s = smem + 3 * ROWS * STRIDE;   // per-row entropy totals [ROWS]

    const int tid  = threadIdx.x;
    const int row0 = blockIdx.x * ROWS;

    if (tid < ROWS) es_s[tid] = 0.0f;
    __syncthreads();

    // ---------------- TDM: values tile -> LDS, padded rows (stride 129) -------------
    if (tid < 32) {  // wave 0 only (TDM ignores EXEC; skip via uniform branch)
        uint64_t ga = (uint64_t)(size_t)(values + (size_t)row0 * T_LEN);
        uint32_t la = (uint32_t)(size_t)v_s;  // LDS byte offset (low 32 bits of flat addr)
        u32x4 g0;
        g0.x = 1u;                             // count=1, user descriptor
        g0.y = la;                             // lds_addr
        g0.z = (uint32_t)ga;                   // global_addr[31:0]
        g0.w = ((uint32_t)(ga >> 32) & 0x01FFFFFFu) | (2u << 30); // addr[56:32], type=2
        i32x8 g1;
        // workgroup_mask=0 | data_size=4B(2)<<16 | pad_enable<<20 | pad_interval=128DW(6)<<22 | pad_amount=1DW(0)<<25
        g1[0] = (2 << 16) | (1 << 20) | (6 << 22);
        g1[1] = (T_LEN & 0xFFFF) << 16;                     // tensor_dim0[15:0] @ bits 63:48
        g1[2] = ((T_LEN >> 16) & 0xFFFF) | ((ROWS & 0xFFFF) << 16); // dim0 hi | tensor_dim1 lo
        g1[3] = ((ROWS >> 16) & 0xFFFF) | ((T_LEN & 0xFFFF) << 16); // dim1 hi | tile_dim0
        g1[4] = (ROWS & 0xFFFF);                            // tile_dim1 | tile_dim2=0 (2D)
        g1[5] = T_LEN;                                      // tensor_dim0_stride[31:0]
        g1[6] = 0;                                          // stride hi | dim1_stride lo
        g1[7] = 0;
        i32x4 g2 = {0, 0, 0, 0};                            // dims 2+ unused (tile_dim2=0)
        i32x4 g3 = {0, 0, 0, 0};
        i32x8 g4 = {0, 0, 0, 0, 0, 0, 0, 0};                // extra group (clang-23 6-arg form)
        __builtin_amdgcn_tensor_load_to_lds(g0, g1, g2, g3, g4, 0);
    }

    // ---------------- async global->LDS: rewards tile, padded rows ------------------
    {
        const float* rb = rewards + (size_t)row0 * T_LEN;
        uint32_t rl = (uint32_t)(size_t)r_s;
        #pragma unroll
        for (int k = 0; k < 4; ++k) {
            int e   = tid + (k << 8);         // float4 index: 0..1023 (32 rows x 32)
            int row = e >> 5;
            int c4  = e & 31;
            uint64_t ga = (uint64_t)(size_t)(rb + (size_t)e * 4);
            uint32_t la = rl + (uint32_t)((row * STRIDE + (c4 << 2)) * 4);
            asm volatile("global_load_async_to_lds_b128 %0, %1, off"
                         :: "v"(la), "v"(ga) : "memory");
        }
    }

    // -------- stream log_probs / entropies (NT b128), reduce A=8 on the fly --------
    {
        const float* lpb = log_probs + (size_t)row0 * T_LEN * A_DIM;
        const float* enb = entropies + (size_t)row0 * T_LEN * A_DIM;
        #pragma unroll 4
        for (int k = 0; k < 16; ++k) {
            int p   = tid + (k << 8);         // (row,t) pair: 0..4095
            int row = p >> 7;
            int t   = p & (T_LEN - 1);
            const v4f* l4 = (const v4f*)(lpb + (size_t)p * A_DIM);
            v4f a0 = __builtin_nontemporal_load(l4);
            v4f a1 = __builtin_nontemporal_load(l4 + 1);
            lp_s[row * STRIDE + t] = (a0.x + a0.y) + (a0.z + a0.w)
                                   + (a1.x + a1.y) + (a1.z + a1.w);
            const v4f* e4 = (const v4f*)(enb + (size_t)p * A_DIM);
            v4f b0 = __builtin_nontemporal_load(e4);
            v4f b1 = __builtin_nontemporal_load(e4 + 1);
            float es = (b0.x + b0.y) + (b0.z + b0.w)
                     + (b1.x + b1.y) + (b1.z + b1.w);
            atomicAdd(&es_s[row], es);        // ds_add_f32
        }
    }

    // -------- drain DMA engines, then workgroup barrier -----------------------------
    asm volatile("s_wait_asynccnt 0x0" ::: "memory");
    __builtin_amdgcn_s_wait_tensorcnt(0);
    __syncthreads();

    // -------- reverse-time scan: one lane per row out of conflict-free LDS ----------
    if (tid < ROWS) {
        const int b = row0 + tid;
        const float R0 = terminal_mask[b] ? 0.0f : last_value[b];
        const int base = tid * STRIDE;
        float R = R0, vn = R0;
        float gae = 0.0f, actor = 0.0f, critic = 0.0f;
        #pragma unroll 8
        for (int t = T_LEN - 1; t >= 0; --t) {
            float r  = r_s[base + t];
            float v  = v_s[base + t];
            float lp = lp_s[base + t];
            R = fmaf(GAMMA, R, r);
            float adv = R - v;
            critic = fmaf(0.5f * adv, adv, critic);
            float delta = fmaf(GAMMA, vn, r) - v;
            gae = fmaf(GAMMA, gae, delta);
            actor = fmaf(-lp, gae, actor);
            vn = v;
        }
        actor -= BETA * es_s[tid];
        out[(size_t)b * 2 + 0] = actor;
        out[(size_t)b * 2 + 1] = critic;
    }
}

extern "C" void kernel_launch(void* const* d_in, const int* in_sizes, int n_in,
                              void* d_out, int out_size, void* d_ws, size_t ws_size,
                              hipStream_t stream) {
    (void)n_in; (void)out_size; (void)d_ws; (void)ws_size;
    const float* values        = (const float*)d_in[0];
    const float* last_value    = (const float*)d_in[1];
    const float* rewards       = (const float*)d_in[2];
    const float* log_probs     = (const float*)d_in[3];
    const float* entropies     = (const float*)d_in[4];
    const int*   terminal_mask = (const int*)  d_in[5];
    float* out = (float*)d_out;

    const int B = in_sizes[1];          // last_value has B elements
    const int grid = B / ROWS;          // 65536/32 = 2048 blocks
    hipLaunchKernelGGL(a3c_scan_kernel, dim3(grid), dim3(THREADS), 0, stream,
                       values, last_value, rewards, log_probs, entropies,
                       terminal_mask, out);
}